// MultiLoss_6579889897517
// MI455X (gfx1250) — compile-verified
//
#include <hip/hip_runtime.h>
#include <hip/hip_bf16.h>
#include <stdint.h>

#define N_COLS        107
#define ROWS_PER_TILE 128
#define BLOCK_THREADS 128

// Per-column classification: 0x80|slot -> MSE column (slot 0..4), else CE slice id 0..7.
__constant__ unsigned char c_kind[N_COLS] = {
    0x80,                                              // col 0        -> mse slot 0
    0,0,0,0,0,0,0,0,0,                                 // cols 1-9     -> slice 0
    1,1,1,1,1,1,1,1,1,1,1,1,1,1,1,1,                   // cols 10-25   -> slice 1
    0x81,                                              // col 26       -> mse slot 1
    2,2,2,2,2,2,2,                                     // cols 27-33   -> slice 2
    3,3,3,3,3,3,3,3,3,3,3,3,3,3,3,                     // cols 34-48   -> slice 3
    4,4,4,4,4,4,                                       // cols 49-54   -> slice 4
    5,5,5,5,5,                                         // cols 55-59   -> slice 5
    6,6,                                               // cols 60-61   -> slice 6
    0x82,0x83,0x84,                                    // cols 62-64   -> mse slots 2-4
    7,7,7,7,7,7,7,7,7,7,7,7,7,7,7,7,7,7,7,7,7,        // cols 65-106  -> slice 7
    7,7,7,7,7,7,7,7,7,7,7,7,7,7,7,7,7,7,7,7,7
};
__constant__ int c_sstart[8] = {1, 10, 27, 34, 49, 55, 60, 65};
__constant__ int c_send[8]   = {10, 26, 34, 49, 55, 60, 62, 107};
__constant__ int c_msecol[5] = {0, 26, 62, 63, 64};

__device__ __forceinline__ void wait_asynccnt0() {
#if defined(__has_builtin) && __has_builtin(__builtin_amdgcn_s_wait_asynccnt)
    __builtin_amdgcn_s_wait_asynccnt(0);
#else
    asm volatile("s_wait_asynccnt 0" ::: "memory");
#endif
}

__global__ __launch_bounds__(BLOCK_THREADS) void
multiloss_tile_kernel(const float* __restrict__ dec,
                      const float* __restrict__ tru,
                      float* __restrict__ partials)
{
    __shared__ __align__(16) float s_dec[ROWS_PER_TILE * N_COLS]; // 54784 B decoded tile
    __shared__ float s_mtrue[ROWS_PER_TILE * 5];                  //  2560 B MSE targets
    __shared__ int   s_label[ROWS_PER_TILE * 8];                  //  4096 B one-hot idx
    __shared__ float s_red[2 * BLOCK_THREADS];                    //  1024 B reduction

    const int    tid        = threadIdx.x;
    const int    tile_elems = ROWS_PER_TILE * N_COLS;             // 13696
    const size_t base       = (size_t)blockIdx.x * (size_t)tile_elems;

    // ---- 1) Fire-and-forget: DMA the decoded tile into LDS (ASYNCcnt path).
    //         Tile byte size 54784 = 3424 x 16B, tile base 16B-aligned.
    {
        const char*    gbase = (const char*)(dec + base);
        const uint32_t lds0  = (uint32_t)(uintptr_t)(&s_dec[0]);
        const int nchunks = (tile_elems * 4) / 16;                // 3424
        for (int i = tid; i < nchunks; i += BLOCK_THREADS) {
            const char*    g = gbase + (size_t)i * 16;
            const uint32_t l = lds0 + (uint32_t)i * 16u;
            asm volatile("global_load_async_to_lds_b128 %0, %1, off"
                         :: "v"(l), "v"(g) : "memory");
        }
    }

    // ---- 2) Init labels (defensive), barrier before scatter writes.
    for (int i = tid; i < ROWS_PER_TILE * 8; i += BLOCK_THREADS) s_label[i] = 0;
    __syncthreads();

    // ---- 3) Coalesced scan of data_true tile while the DMA runs:
    //         pull out 8 one-hot column indices + 5 MSE targets per row.
    for (int i = tid; i < tile_elems; i += BLOCK_THREADS) {
        float v = tru[base + (size_t)i];
        int row = (int)((unsigned)i / (unsigned)N_COLS);
        int col = i - row * N_COLS;
        unsigned k = c_kind[col];
        if (k & 0x80u) {
            s_mtrue[row * 5 + (int)(k & 7u)] = v;
        } else if (v > 0.5f) {                  // exact one-hot -> argmax position
            s_label[row * 8 + (int)k] = col - c_sstart[k];
        }
    }

    // ---- 4) Drain async DMA, make tile visible block-wide.
    wait_asynccnt0();
    __syncthreads();

    // ---- 5) One row per thread, straight from LDS (conflict-free: stride 107 dwords).
    const float* r = &s_dec[tid * N_COLS];

    float mse_acc = 0.0f;
#pragma unroll
    for (int k = 0; k < 5; ++k) {
        float d = r[c_msecol[k]] - s_mtrue[tid * 5 + k];
        mse_acc += d * d;
    }

    float ce_acc = 0.0f;
#pragma unroll
    for (int j = 0; j < 8; ++j) {
        const int s = c_sstart[j], e = c_send[j];
        float m = -3.402823466e38f;
        for (int c = s; c < e; ++c) m = fmaxf(m, r[c]);
        float sum = 0.0f;
        for (int c = s; c < e; ++c) sum += __expf(r[c] - m);
        const int lbl = s_label[tid * 8 + j];
        ce_acc += m + __logf(sum) - r[s + lbl];  // = -log_softmax[label]
    }

    // ---- 6) Deterministic block tree reduction -> per-block partials.
    s_red[tid]                 = mse_acc;
    s_red[BLOCK_THREADS + tid] = ce_acc;
    __syncthreads();
    for (int off = BLOCK_THREADS / 2; off > 0; off >>= 1) {
        if (tid < off) {
            s_red[tid]                 += s_red[tid + off];
            s_red[BLOCK_THREADS + tid] += s_red[BLOCK_THREADS + tid + off];
        }
        __syncthreads();
    }
    if (tid == 0) {
        partials[(size_t)blockIdx.x * 2 + 0] = s_red[0];
        partials[(size_t)blockIdx.x * 2 + 1] = s_red[BLOCK_THREADS];
    }
}

__global__ __launch_bounds__(256) void
multiloss_reduce_kernel(const float* __restrict__ partials, int n_tiles,
                        float inv_batch, float* __restrict__ out)
{
    __shared__ float s_m[256];
    __shared__ float s_c[256];
    const int tid = threadIdx.x;
    float m = 0.0f, c = 0.0f;
    for (int i = tid; i < n_tiles; i += 256) {   // fixed order -> deterministic
        m += partials[(size_t)i * 2 + 0];
        c += partials[(size_t)i * 2 + 1];
    }
    s_m[tid] = m; s_c[tid] = c;
    __syncthreads();
    for (int off = 128; off > 0; off >>= 1) {
        if (tid < off) { s_m[tid] += s_m[tid + off]; s_c[tid] += s_c[tid + off]; }
        __syncthreads();
    }
    if (tid == 0) {
        float mse = s_m[0] * inv_batch;
        float ce  = s_c[0] * inv_batch;
        out[0] = mse + ce;   // multi_loss
        out[1] = mse;        // mse_loss
        out[2] = ce;         // ce_loss
    }
}

extern "C" void kernel_launch(void* const* d_in, const int* in_sizes, int n_in,
                              void* d_out, int out_size, void* d_ws, size_t ws_size,
                              hipStream_t stream) {
    const float* dec = (const float*)d_in[0];   // data_decoded, [batch, 107] f32
    const float* tru = (const float*)d_in[1];   // data_true,    [batch, 107] f32
    float* out      = (float*)d_out;            // 3 f32 scalars
    float* partials = (float*)d_ws;             // n_tiles * 2 f32

    const long long total = (long long)in_sizes[0];
    const int batch   = (int)(total / N_COLS);          // 524288
    const int n_tiles = batch / ROWS_PER_TILE;          // 4096

    multiloss_tile_kernel<<<n_tiles, BLOCK_THREADS, 0, stream>>>(dec, tru, partials);
    multiloss_reduce_kernel<<<1, 256, 0, stream>>>(partials, n_tiles,
                                                   1.0f / (float)batch, out);
}